// input_timestep_5798205849688
// MI455X (gfx1250) — compile-verified
//
#include <hip/hip_runtime.h>
#include <hip/hip_bf16.h>
#include <math.h>

typedef __attribute__((ext_vector_type(2))) float v2f;
typedef __attribute__((ext_vector_type(4))) float v4f;
typedef __attribute__((ext_vector_type(8))) float v8f;

#define EMB 128
#define LDSTR 132   // padded LDS row stride (floats) to avoid bank conflicts

// ---------------------------------------------------------------------------
// Kernel 1: sinusoidal embedding + 2x (Linear + SiLU) via V_WMMA_F32_16X16X4_F32
// One wave (32 lanes) per block handles a 16-row tile of the batch.
// ---------------------------------------------------------------------------
__global__ __launch_bounds__(32) void mlp_wmma_kernel(
    const float* __restrict__ t,
    const float* __restrict__ W1, const float* __restrict__ b1,
    const float* __restrict__ W2, const float* __restrict__ b2,
    float* __restrict__ h_out, int Btot)
{
    __shared__ float lemb[16 * LDSTR];  // embedding tile  [16][128] (padded)
    __shared__ float lh[16 * LDSTR];    // hidden tile     [16][128] (padded)

    const int lane = threadIdx.x;       // 0..31 (wave32)
    const int l15  = lane & 15;
    const int lh16 = lane >> 4;         // 0 or 1: selects K pair / M half
    const int m0   = blockIdx.x * 16;

    // --- sinusoidal embedding: emb[row][col] ; col<64 -> sin, else cos -----
    // freqs = exp(linspace(ln 1, ln 1000, 64)); ang = 2*pi*freq
    const float LNSTEP = 0.1096469090f;           // ln(1000)/63
    for (int idx = lane; idx < 16 * EMB; idx += 32) {
        int row = idx >> 7;
        int col = idx & 127;
        int g   = m0 + row;
        float tv = (g < Btot) ? t[g] : 0.0f;
        int f = col & 63;
        float ang = 6.28318530717958647f * expf(LNSTEP * (float)f);
        float ph  = tv * ang;
        lemb[row * LDSTR + col] = (col < 64) ? sinf(ph) : cosf(ph);
    }
    __syncthreads();

    // --- layer 1: h1 = silu(emb @ W1^T + b1) ------------------------------
    // D[m][n] = sum_k A[m][k] * B[k][n],  A = emb tile, B[k][n] = W1[n][k]
    v8f acc[8];
    for (int nt = 0; nt < 8; ++nt) {
        float bv = b1[nt * 16 + l15];           // C[m][n] = b1[n], all m
        for (int r = 0; r < 8; ++r) acc[nt][r] = bv;
    }
    for (int k0 = 0; k0 < EMB; k0 += 4) {
        // A 16x4 f32 fragment (2 VGPRs): lanes0-15 K={k0,k0+1}, lanes16-31 K={k0+2,k0+3}
        v2f a;
        a.x = lemb[l15 * LDSTR + k0 + 2 * lh16];
        a.y = lemb[l15 * LDSTR + k0 + 2 * lh16 + 1];
        for (int nt = 0; nt < 8; ++nt) {
            // B 4x16 fragment: lane holds N = l15, K pair selected by lh16
            const float* wr = W1 + (size_t)(nt * 16 + l15) * EMB + k0 + 2 * lh16;
            v2f b; b.x = wr[0]; b.y = wr[1];
            acc[nt] = __builtin_amdgcn_wmma_f32_16x16x4_f32(
                false, a, false, b, (short)0, acc[nt], false, false);
        }
    }

    // SiLU, scatter D fragments back to LDS as the next layer's A operand
    for (int nt = 0; nt < 8; ++nt) {
        for (int r = 0; r < 8; ++r) {
            float x = acc[nt][r];
            float s = x / (1.0f + __expf(-x));
            int row = r + 8 * lh16;             // D layout: lanes16-31 -> M = r+8
            lh[row * LDSTR + nt * 16 + l15] = s;
        }
    }
    __syncthreads();

    // --- layer 2: h2 = silu(h1 @ W2^T + b2) -------------------------------
    for (int nt = 0; nt < 8; ++nt) {
        float bv = b2[nt * 16 + l15];
        for (int r = 0; r < 8; ++r) acc[nt][r] = bv;
    }
    for (int k0 = 0; k0 < EMB; k0 += 4) {
        v2f a;
        a.x = lh[l15 * LDSTR + k0 + 2 * lh16];
        a.y = lh[l15 * LDSTR + k0 + 2 * lh16 + 1];
        for (int nt = 0; nt < 8; ++nt) {
            const float* wr = W2 + (size_t)(nt * 16 + l15) * EMB + k0 + 2 * lh16;
            v2f b; b.x = wr[0]; b.y = wr[1];
            acc[nt] = __builtin_amdgcn_wmma_f32_16x16x4_f32(
                false, a, false, b, (short)0, acc[nt], false, false);
        }
    }

    // SiLU + write h[8192][128] to workspace
    for (int nt = 0; nt < 8; ++nt) {
        for (int r = 0; r < 8; ++r) {
            float x = acc[nt][r];
            float s = x / (1.0f + __expf(-x));
            int row = r + 8 * lh16;
            int g   = m0 + row;
            if (g < Btot) h_out[(size_t)g * EMB + nt * 16 + l15] = s;
        }
    }
}

// ---------------------------------------------------------------------------
// Kernel 2: broadcast h[b][:] -> out[b][0][0:128][:]  (512 MB, store-bound)
// Each thread loads its 16B slice of h once, stores it 16x with NT hint.
// ---------------------------------------------------------------------------
__global__ __launch_bounds__(256) void broadcast_kernel(
    const float* __restrict__ h, float* __restrict__ out)
{
    const int b   = blockIdx.x;
    const int tid = threadIdx.x;
    const int w0  = (tid * 4) & 127;              // column this float4 covers
    v4f v = *(const v4f*)(h + (size_t)b * EMB + w0);
    float* base = out + (size_t)b * (EMB * EMB) + tid * 4;
    #pragma unroll
    for (int it = 0; it < 16; ++it) {
        __builtin_nontemporal_store(v, (v4f*)(base + it * 1024));
    }
}

extern "C" void kernel_launch(void* const* d_in, const int* in_sizes, int n_in,
                              void* d_out, int out_size, void* d_ws, size_t ws_size,
                              hipStream_t stream) {
    const float* t  = (const float*)d_in[0];
    const float* W1 = (const float*)d_in[1];
    const float* b1 = (const float*)d_in[2];
    const float* W2 = (const float*)d_in[3];
    const float* b2 = (const float*)d_in[4];
    float* out = (float*)d_out;
    float* h   = (float*)d_ws;                    // 8192*128*4 = 4 MB scratch
    const int B = in_sizes[0];                    // 8192

    mlp_wmma_kernel<<<(B + 15) / 16, 32, 0, stream>>>(t, W1, b1, W2, b2, h, B);
    broadcast_kernel<<<B, 256, 0, stream>>>(h, out);
}